// Circuit_31662498906421
// MI455X (gfx1250) — compile-verified
//
#include <hip/hip_runtime.h>
#include <math.h>

// ---------------------------------------------------------------------------
// 16-qubit, 2-layer parameterized circuit, batch 32 (state = 16 MB, L2-resident).
//   * fuse Rz*Ry*Rx per qubit -> 2x2 complex U (rotations commute across qubits)
//   * kron-fuse 4 qubits -> 16x16 complex gate (8 total: 2 layers x 4 groups)
//   * apply TWO 16x16 gates per pass: GEMM on nibble A, transpose the register
//     tile through wave-private LDS, GEMM on nibble B  -> 4 full-state passes
//   * CNOT ring is a FIXED GF(2)-linear index permutation -> folded into the
//     stores of each layer's last pass via compile-time bit-matrix constants
//   * |psi|^2 folded into the final pass (stores probs directly to d_out)
// ---------------------------------------------------------------------------

typedef float v2f __attribute__((ext_vector_type(2)));
typedef float v8f __attribute__((ext_vector_type(8)));

#define DIM   65536
#define BATCH 32
#define NST   (BATCH * DIM)      // floats per state component
#define TPW   4                  // tiles per wave
#define LROW  20                 // LDS row stride (floats), banking: 5r mod 16

// Forward CNOT-ring permutation of a 16-bit basis index (qubit q <-> bit 15-q):
// for q=0..15: bit(15-((q+1)&15)) ^= bit(15-q).  GF(2)-linear, parameter-free.
__host__ __device__ constexpr unsigned fwd16(unsigned x) {
    for (int q = 0; q < 16; ++q)
        x ^= ((x >> (15 - q)) & 1u) << (15 - ((q + 1) & 15));
    return x;
}

// -------------------- gate preparation -------------------------------------
__global__ __launch_bounds__(256)
void prep_gates(const float* __restrict__ thetas,
                float* __restrict__ gre, float* __restrict__ gim,
                float* __restrict__ gng) {
    const int lg = blockIdx.x;
    const int l = lg >> 2, g = lg & 3;
    const int tid = threadIdx.x;
    const int j = tid >> 4, m = tid & 15;

    float ar = 1.f, ai = 0.f;
#pragma unroll
    for (int k = 0; k < 4; ++k) {
        const int q = 4 * g + k;
        const float tx = thetas[l * 48 + q];
        const float ty = thetas[l * 48 + 16 + q];
        const float tz = thetas[l * 48 + 32 + q];
        const float cx = cosf(0.5f * tx), sx = sinf(0.5f * tx);
        const float cy = cosf(0.5f * ty), sy = sinf(0.5f * ty);
        const float cz = cosf(0.5f * tz), sz = sinf(0.5f * tz);
        float er[2][2], ei[2][2];                    // Ry*Rx
        er[0][0] =  cy * cx;  ei[0][0] =  sy * sx;
        er[0][1] = -sy * cx;  ei[0][1] = -cy * sx;
        er[1][0] =  sy * cx;  ei[1][0] = -cy * sx;
        er[1][1] =  cy * cx;  ei[1][1] = -sy * sx;
#pragma unroll
        for (int row = 0; row < 2; ++row) {          // Rz * (Ry*Rx)
            const float pr = cz, pi = (row == 0) ? -sz : sz;
#pragma unroll
            for (int col = 0; col < 2; ++col) {
                const float rr = er[row][col] * pr - ei[row][col] * pi;
                const float ii = er[row][col] * pi + ei[row][col] * pr;
                er[row][col] = rr; ei[row][col] = ii;
            }
        }
        const int jb = (j >> (3 - k)) & 1;
        const int mb = (m >> (3 - k)) & 1;
        const float ur = er[jb][mb], ui = ei[jb][mb];
        const float nr = ar * ur - ai * ui;
        const float ni = ar * ui + ai * ur;
        ar = nr; ai = ni;
    }
    const int off = lg * 256 + j * 16 + m;
    gre[off] = ar;
    gim[off] = ai;
    gng[off] = -ai;   // pre-negated imag: f32 WMMA NEG modifies C only (ISA 7.12)
}

// -------------------- two fused 16x16 complex gates via WMMA ----------------
// PA / PB: bit positions of the two nibbles (gate A at PA, gate B at PB).
// RESTSHIFT: bit position of the 8 untouched index bits.
// OUTMODE 0: normal store; 1: CNOT-permuted scatter (PA=4,PB=0);
//         2: CNOT-permuted scatter of |amp|^2 to probs output.
template<int PA, int PB, int RESTSHIFT, int OUTMODE>
__global__ __launch_bounds__(256)
void apply_pair(const float* __restrict__ inre, const float* __restrict__ inim,
                float* __restrict__ outre, float* __restrict__ outim,
                const float* __restrict__ gAre, const float* __restrict__ gAim,
                const float* __restrict__ gAng,
                const float* __restrict__ gBre, const float* __restrict__ gBim,
                const float* __restrict__ gBng) {
    static_assert(OUTMODE == 0 || (PA == 4 && PB == 0), "perm store needs low nibbles");

    __shared__ float sre[8][16 * LROW];   // wave-private transpose staging
    __shared__ float sim_[8][16 * LROW];

    const int lane = threadIdx.x & 31;
    const int wave = threadIdx.x >> 5;
    const int r  = lane & 15;    // N index of B/D operands, M index of A operand
    const int hi = lane >> 4;    // K-half selector (ISA f32 A/B VGPR layout)

    // ---- both gates as four 16x4 A-slices each (re, im, -im) ----
    v2f Are[4], Aim[4], Ang[4], Bre_[4], Bim_[4], Bng_[4];
#pragma unroll
    for (int s = 0; s < 4; ++s) {
        const int o = r * 16 + 4 * s + 2 * hi;
        Are[s].x  = gAre[o]; Are[s].y  = gAre[o + 1];
        Aim[s].x  = gAim[o]; Aim[s].y  = gAim[o + 1];
        Ang[s].x  = gAng[o]; Ang[s].y  = gAng[o + 1];
        Bre_[s].x = gBre[o]; Bre_[s].y = gBre[o + 1];
        Bim_[s].x = gBim[o]; Bim_[s].y = gBim[o + 1];
        Bng_[s].x = gBng[o]; Bng_[s].y = gBng[o + 1];
    }

    // compile-time CNOT-permutation images of the output-nibble bits
    const unsigned FJ[8] = { fwd16(0u), fwd16(1u), fwd16(2u), fwd16(3u),
                             fwd16(4u), fwd16(5u), fwd16(6u), fwd16(7u) };
    const unsigned FH = fwd16(8u);
    const unsigned C4 = fwd16(1u << 4), C5 = fwd16(2u << 4),
                   C6 = fwd16(4u << 4), C7 = fwd16(8u << 4);

    float* __restrict__ lr = sre[wave];
    float* __restrict__ li = sim_[wave];

    const int wid = blockIdx.x * 8 + wave;            // 2048 waves x 4 tiles
    for (int it = 0; it < TPW; ++it) {
        const int t13  = wid * TPW + it;              // (batch, rest8) tile id
        const int b    = t13 >> 8;
        const int rest = t13 & 255;
        const int base = (b << 16) + (rest << RESTSHIFT);

        // ---- GEMM 1: D[jA, mB] = sum_mA GA[jA,mA] * X[mA, mB]  (rows = mB) --
        v8f dre = {}, dim_ = {};
#pragma unroll
        for (int s = 0; s < 4; ++s) {
            const int m0 = 4 * s + 2 * hi;
            const int o0 = base + (m0 << PA) + (r << PB);
            const int o1 = base + ((m0 + 1) << PA) + (r << PB);
            v2f xre, xim;
            xre.x = inre[o0]; xre.y = inre[o1];
            xim.x = inim[o0]; xim.y = inim[o1];
            dre  = __builtin_amdgcn_wmma_f32_16x16x4_f32(false, Are[s], false, xre,
                                                         (short)0, dre,  false, false);
            dre  = __builtin_amdgcn_wmma_f32_16x16x4_f32(false, Ang[s], false, xim,
                                                         (short)0, dre,  false, false);
            dim_ = __builtin_amdgcn_wmma_f32_16x16x4_f32(false, Aim[s], false, xre,
                                                         (short)0, dim_, false, false);
            dim_ = __builtin_amdgcn_wmma_f32_16x16x4_f32(false, Are[s], false, xim,
                                                         (short)0, dim_, false, false);
        }

        // ---- transpose D through wave-private LDS: [mB][jA] ----------------
#pragma unroll
        for (int v = 0; v < 8; ++v) {
            lr[r * LROW + v + 8 * hi] = dre[v];
            li[r * LROW + v + 8 * hi] = dim_[v];
        }

        // ---- GEMM 2: E[jB, jA] = sum_mB GB[jB,mB] * D[jA, mB] --------------
        v8f ere = {}, eim = {};
#pragma unroll
        for (int s = 0; s < 4; ++s) {
            const int k0 = 4 * s + 2 * hi;
            v2f tre, tim;
            tre.x = lr[k0 * LROW + r];  tre.y = lr[(k0 + 1) * LROW + r];
            tim.x = li[k0 * LROW + r];  tim.y = li[(k0 + 1) * LROW + r];
            ere = __builtin_amdgcn_wmma_f32_16x16x4_f32(false, Bre_[s], false, tre,
                                                        (short)0, ere, false, false);
            ere = __builtin_amdgcn_wmma_f32_16x16x4_f32(false, Bng_[s], false, tim,
                                                        (short)0, ere, false, false);
            eim = __builtin_amdgcn_wmma_f32_16x16x4_f32(false, Bim_[s], false, tre,
                                                        (short)0, eim, false, false);
            eim = __builtin_amdgcn_wmma_f32_16x16x4_f32(false, Bre_[s], false, tim,
                                                        (short)0, eim, false, false);
        }

        // ---- store: lane holds jA = r, register v holds jB = v + 8*hi ------
        if (OUTMODE == 0) {
#pragma unroll
            for (int v = 0; v < 8; ++v) {
                const int o = base + (r << PA) + ((v + 8 * hi) << PB);
                outre[o] = ere[v];
                outim[o] = eim[v];
            }
        } else {
            // index = base16 ^ (r<<4) ^ jB over disjoint bit fields; fwd16 is
            // GF(2)-linear -> permuted address XOR-decomposes into constants.
            unsigned fb = (unsigned)base & 0xFFFFu;     // wave-uniform
#pragma unroll
            for (int q = 0; q < 16; ++q)
                fb ^= ((fb >> (15 - q)) & 1u) << (15 - ((q + 1) & 15));
            const unsigned fa = ((r & 1) ? C4 : 0u) ^ ((r & 2) ? C5 : 0u) ^
                                ((r & 4) ? C6 : 0u) ^ ((r & 8) ? C7 : 0u);
            const unsigned bb = (unsigned)base & 0xFFFF0000u;
            const unsigned fh = hi ? FH : 0u;
#pragma unroll
            for (int v = 0; v < 8; ++v) {
                const unsigned o = bb | (fb ^ fa ^ fh ^ FJ[v]);
                if (OUTMODE == 1) {
                    outre[o] = ere[v];
                    outim[o] = eim[v];
                } else {
                    outre[o] = ere[v] * ere[v] + eim[v] * eim[v];  // probs
                }
            }
        }
    }
}

// -------------------- launcher ---------------------------------------------
extern "C" void kernel_launch(void* const* d_in, const int* in_sizes, int n_in,
                              void* d_out, int out_size, void* d_ws, size_t ws_size,
                              hipStream_t stream) {
    (void)in_sizes; (void)n_in; (void)out_size; (void)ws_size;
    const float* thetas = (const float*)d_in[0];   // (2,3,16)
    const float* in_re  = (const float*)d_in[1];   // (32, 65536)
    const float* in_im  = (const float*)d_in[2];

    float* ws  = (float*)d_ws;
    float* GRE = ws;                 // 8 * 256 gate tables
    float* GIM = ws + 2048;
    float* GNG = ws + 4096;
    float* Are = ws + 8192;          // ping-pong state (SoA re/im)
    float* Aim = Are + NST;
    float* Bre = Aim + NST;
    float* Bim = Bre + NST;

    prep_gates<<<8, 256, 0, stream>>>(thetas, GRE, GIM, GNG);

    const dim3 grid(256), blk(256);  // 2048 waves x 4 tiles = 8192 tiles/pass

#define G(lg) GRE + (lg) * 256, GIM + (lg) * 256, GNG + (lg) * 256

    // layer 0: qubits 0-7 (bits 15-8), then qubits 8-15 (bits 7-0) + CNOT perm
    apply_pair<12, 8, 0, 0><<<grid, blk, 0, stream>>>(in_re, in_im, Are, Aim,
                                                      G(0), G(1));
    apply_pair< 4, 0, 8, 1><<<grid, blk, 0, stream>>>(Are, Aim, Bre, Bim,
                                                      G(2), G(3));
    // layer 1: same, final pass writes |psi|^2 straight to d_out
    apply_pair<12, 8, 0, 0><<<grid, blk, 0, stream>>>(Bre, Bim, Are, Aim,
                                                      G(4), G(5));
    apply_pair< 4, 0, 8, 2><<<grid, blk, 0, stream>>>(Are, Aim, (float*)d_out, nullptr,
                                                      G(6), G(7));
#undef G
}